// BaselineModel_70901320122674
// MI455X (gfx1250) — compile-verified
//
#include <hip/hip_runtime.h>
#include <math.h>

// ---------------------------------------------------------------------------
// Types for CDNA5 WMMA (wave32).  V_WMMA_F32_16X16X4_F32:
//   A: 16x4 f32 (2 VGPRs/lane), B: 4x16 f32 (2 VGPRs/lane), C/D: 16x16 f32 (8 VGPRs/lane)
// ---------------------------------------------------------------------------
typedef __attribute__((ext_vector_type(2))) float v2f;
typedef __attribute__((ext_vector_type(8))) float v8f;

#define HDIM 32
#define NEG_SLOPE 0.2f

// ---------------------------------------------------------------------------
// Generic tall-skinny GEMM: Out[r, c] = sum_k act(X[r,k]) * W[c,k]  (+ bias[c])
// K fixed at 32, ncols in {32, 96}. One wave handles 16 rows; A fragments are
// reused across all 16-wide column tiles.  Full f32 via v_wmma_f32_16x16x4_f32.
// ---------------------------------------------------------------------------
__global__ void gemm_xwt_kernel(const float* __restrict__ X,
                                const float* __restrict__ W,
                                const float* __restrict__ bias,
                                float* __restrict__ Out,
                                int ngroups, int ncols, int relu)
{
    const int wave = blockIdx.x * (blockDim.x >> 5) + (threadIdx.x >> 5);
    if (wave >= ngroups) return;                 // wave-uniform exit: EXEC stays all-ones
    const int lane = threadIdx.x & 31;
    const int half = lane >> 4;                  // 0: K pair (0,1); 1: K pair (2,3)
    const int lrow = lane & 15;
    const int base = wave * 16;

    // ---- A fragments: row = base+lrow, per chunk c covers K = 4c..4c+3 ----
    v2f a[8];
    const float* xrow = X + (size_t)(base + lrow) * HDIM;
    #pragma unroll
    for (int c = 0; c < 8; ++c) {
        const int k = 4 * c + 2 * half;
        float x0 = xrow[k];
        float x1 = xrow[k + 1];
        if (relu) { x0 = fmaxf(x0, 0.0f); x1 = fmaxf(x1, 0.0f); }
        v2f av; av.x = x0; av.y = x1;
        a[c] = av;
    }

    for (int j0 = 0; j0 < ncols; j0 += 16) {
        const int col = j0 + lrow;
        const float* wrow = W + (size_t)col * HDIM;   // B[k][col] = W[col][k]
        v8f acc = {};
        #pragma unroll
        for (int c = 0; c < 8; ++c) {
            const int k = 4 * c + 2 * half;
            v2f bv; bv.x = wrow[k]; bv.y = wrow[k + 1];
            acc = __builtin_amdgcn_wmma_f32_16x16x4_f32(
                /*neg_a=*/false, a[c], /*neg_b=*/false, bv,
                /*c_mod=*/(short)0, acc, /*reuse_a=*/false, /*reuse_b=*/false);
        }
        const float badd = bias ? bias[col] : 0.0f;
        #pragma unroll
        for (int v = 0; v < 8; ++v) {
            const int m = v + (half ? 8 : 0);         // C/D layout: rows v / v+8
            Out[(size_t)(base + m) * ncols + col] = acc[v] + badd;
        }
    }
}

// ---------------------------------------------------------------------------
// Order-preserving float <-> uint map for atomicMax-based segment max.
// ---------------------------------------------------------------------------
__device__ __forceinline__ unsigned int floatFlip(float f) {
    unsigned int u = __float_as_uint(f);
    return (u & 0x80000000u) ? ~u : (u | 0x80000000u);
}
__device__ __forceinline__ float floatUnflip(unsigned int u) {
    return (u & 0x80000000u) ? __uint_as_float(u ^ 0x80000000u)
                             : __uint_as_float(~u);
}
#define NEG_INF_FLIPPED 0x007FFFFFu   // floatFlip(-inf)

__device__ __forceinline__ float sigmoidf_dev(float x) {
    return 1.0f / (1.0f + expf(-x));
}

// per-node attention scalars: a_s = h . att_src, a_d = h . att_dst
__global__ void node_dots_kernel(const float* __restrict__ h,
                                 const float* __restrict__ att_s,
                                 const float* __restrict__ att_d,
                                 float* __restrict__ a_s,
                                 float* __restrict__ a_d, int N)
{
    int t = blockIdx.x * blockDim.x + threadIdx.x;
    if (t >= N) return;
    const float* hr = h + (size_t)t * HDIM;
    float s = 0.0f, d = 0.0f;
    #pragma unroll
    for (int k = 0; k < HDIM; ++k) {
        float hv = hr[k];
        s += hv * att_s[k];
        d += hv * att_d[k];
    }
    a_s[t] = s;
    a_d[t] = d;
}

// zero aggregation buffer, init segment max/denom
__global__ void gat_init_kernel(float* __restrict__ hout,
                                unsigned int* __restrict__ m_u,
                                float* __restrict__ denom, int N)
{
    int t = blockIdx.x * blockDim.x + threadIdx.x;
    if (t < N * HDIM) hout[t] = 0.0f;
    if (t < N) { m_u[t] = NEG_INF_FLIPPED; denom[t] = 0.0f; }
}

// pass 1: e = leaky_relu(a_s[src] + a_d[dst]); segment max into m_u[dst]
__global__ void edge_score_kernel(const int* __restrict__ ei,
                                  const float* __restrict__ a_s,
                                  const float* __restrict__ a_d,
                                  float* __restrict__ e_buf,
                                  unsigned int* __restrict__ m_u,
                                  int E, int N)
{
    int t = blockIdx.x * blockDim.x + threadIdx.x;
    if (t >= E + N) return;
    const int src = (t < E) ? ei[t]     : (t - E);   // self-loop tail
    const int dst = (t < E) ? ei[E + t] : (t - E);
    const float v = a_s[src] + a_d[dst];
    const float e = (v > 0.0f) ? v : NEG_SLOPE * v;
    e_buf[t] = e;
    atomicMax(m_u + dst, floatFlip(e));
}

// pass 2: p = exp(e - m[dst]); denom[dst] += p
__global__ void edge_exp_kernel(const int* __restrict__ ei,
                                float* __restrict__ e_buf,
                                const unsigned int* __restrict__ m_u,
                                float* __restrict__ denom,
                                int E, int N)
{
    int t = blockIdx.x * blockDim.x + threadIdx.x;
    if (t >= E + N) return;
    const int dst = (t < E) ? ei[E + t] : (t - E);
    const float p = expf(e_buf[t] - floatUnflip(m_u[dst]));
    e_buf[t] = p;
    atomicAdd(denom + dst, p);
}

// pass 3: hout[dst] += (p/denom[dst]) * h[src]; lane == feature (coalesced)
__global__ void edge_agg_kernel(const int* __restrict__ ei,
                                const float* __restrict__ e_buf,
                                const float* __restrict__ denom,
                                const float* __restrict__ hsrc,
                                float* __restrict__ hout,
                                int E, int N)
{
    int t = blockIdx.x * blockDim.x + threadIdx.x;
    const int edge = t >> 5;
    const int k = t & 31;
    if (edge >= E + N) return;
    const int src = (edge < E) ? ei[edge]     : (edge - E);
    const int dst = (edge < E) ? ei[E + edge] : (edge - E);
    const float alpha = e_buf[edge] / denom[dst];
    atomicAdd(hout + (size_t)dst * HDIM + k, alpha * hsrc[(size_t)src * HDIM + k]);
}

__global__ void add_bias_kernel(float* __restrict__ buf,
                                const float* __restrict__ bias, int N)
{
    int t = blockIdx.x * blockDim.x + threadIdx.x;
    if (t < N * HDIM) buf[t] += bias[t & 31];
}

// GRU with h_prev = 0: gi already has bih folded in; gh = bhh (constant)
__global__ void gru_gates_kernel(const float* __restrict__ gi,
                                 const float* __restrict__ bhh,
                                 float* __restrict__ hout, int N)
{
    int t = blockIdx.x * blockDim.x + threadIdx.x;
    if (t >= N * HDIM) return;
    const int n = t >> 5;
    const int k = t & 31;
    const float* g = gi + (size_t)n * (3 * HDIM);
    const float r  = sigmoidf_dev(g[k] + bhh[k]);
    const float z  = sigmoidf_dev(g[HDIM + k] + bhh[HDIM + k]);
    const float nn = tanhf(g[2 * HDIM + k] + r * bhh[2 * HDIM + k]);
    hout[t] = (1.0f - z) * nn;
}

__global__ void final_lin_kernel(const float* __restrict__ h,
                                 const float* __restrict__ linW,
                                 const float* __restrict__ linb,
                                 float* __restrict__ out, int N)
{
    int t = blockIdx.x * blockDim.x + threadIdx.x;
    if (t >= N) return;
    const float* hr = h + (size_t)t * HDIM;
    float s = linb[0];
    #pragma unroll
    for (int k = 0; k < HDIM; ++k) s += hr[k] * linW[k];
    out[t] = s;
}

// ---------------------------------------------------------------------------
// Host launch
// ---------------------------------------------------------------------------
extern "C" void kernel_launch(void* const* d_in, const int* in_sizes, int n_in,
                              void* d_out, int out_size, void* d_ws, size_t ws_size,
                              hipStream_t stream)
{
    const int N  = in_sizes[0] / HDIM;      // 100000
    const int E  = in_sizes[1] / 2;         // 2400000
    const int EL = E + N;                   // edges incl. self-loops

    const float* x        = (const float*)d_in[0];
    const int*   ei       = (const int*)  d_in[1];
    const float* conv_W   = (const float*)d_in[2];
    const float* att_src  = (const float*)d_in[3];
    const float* att_dst  = (const float*)d_in[4];
    const float* conv_b   = (const float*)d_in[5];
    const float* Wih0     = (const float*)d_in[6];
    const float* bih0     = (const float*)d_in[8];
    const float* bhh0     = (const float*)d_in[9];
    const float* Wih1     = (const float*)d_in[10];
    const float* bih1     = (const float*)d_in[12];
    const float* bhh1     = (const float*)d_in[13];
    const float* linW     = (const float*)d_in[14];
    const float* linb     = (const float*)d_in[15];
    float* out = (float*)d_out;

    // workspace partition (floats)
    float* ws    = (float*)d_ws;
    float* bufA  = ws;                               // N*32
    float* bufB  = bufA + (size_t)N * HDIM;          // N*32
    float* gi    = bufB + (size_t)N * HDIM;          // N*96
    float* a_s   = gi + (size_t)N * 3 * HDIM;        // N
    float* a_d   = a_s + N;                          // N
    unsigned int* m_u = (unsigned int*)(a_d + N);    // N
    float* denom = (float*)(m_u + N);                // N
    float* e_buf = denom + N;                        // E+N

    auto cdiv = [](int a, int b) { return (a + b - 1) / b; };
    const int TB = 256;
    const int ngroups = N / 16;                      // 6250 exact 16-row tiles
    const int gemmBlocks = cdiv(ngroups, TB / 32);   // 8 waves per block

    auto run_gat_edges = [&](const float* h, float* agg) {
        node_dots_kernel <<<cdiv(N, TB), TB, 0, stream>>>(h, att_src, att_dst, a_s, a_d, N);
        gat_init_kernel  <<<cdiv(N * HDIM, TB), TB, 0, stream>>>(agg, m_u, denom, N);
        edge_score_kernel<<<cdiv(EL, TB), TB, 0, stream>>>(ei, a_s, a_d, e_buf, m_u, E, N);
        edge_exp_kernel  <<<cdiv(EL, TB), TB, 0, stream>>>(ei, e_buf, m_u, denom, E, N);
        edge_agg_kernel  <<<cdiv(EL * 32, TB), TB, 0, stream>>>(ei, e_buf, denom, h, agg, E, N);
        add_bias_kernel  <<<cdiv(N * HDIM, TB), TB, 0, stream>>>(agg, conv_b, N);
    };

    // --- GAT layer 1: h = relu(x) @ W^T, then attention aggregation ---
    gemm_xwt_kernel<<<gemmBlocks, TB, 0, stream>>>(x, conv_W, nullptr, bufA, ngroups, HDIM, 1);
    run_gat_edges(bufA, bufB);

    // --- GRU cell 0 (h0 = 0) ---
    gemm_xwt_kernel<<<gemmBlocks, TB, 0, stream>>>(bufB, Wih0, bih0, gi, ngroups, 3 * HDIM, 0);
    gru_gates_kernel<<<cdiv(N * HDIM, TB), TB, 0, stream>>>(gi, bhh0, bufA, N);

    // --- GRU cell 1 (h0 = 0) ---
    gemm_xwt_kernel<<<gemmBlocks, TB, 0, stream>>>(bufA, Wih1, bih1, gi, ngroups, 3 * HDIM, 0);
    gru_gates_kernel<<<cdiv(N * HDIM, TB), TB, 0, stream>>>(gi, bhh1, bufB, N);

    // --- GAT layer 2 (no relu on input) ---
    gemm_xwt_kernel<<<gemmBlocks, TB, 0, stream>>>(bufB, conv_W, nullptr, bufA, ngroups, HDIM, 0);
    run_gat_edges(bufA, bufB);

    // --- final linear head ---
    final_lin_kernel<<<cdiv(N, TB), TB, 0, stream>>>(bufB, linW, linb, out, N);
}